// GCNLayer_42296837931707
// MI455X (gfx1250) — compile-verified
//
#include <hip/hip_runtime.h>

typedef float v2f __attribute__((ext_vector_type(2)));
typedef float v4f __attribute__((ext_vector_type(4)));
typedef float v8f __attribute__((ext_vector_type(8)));

#define N_NODES 100000
#define N_EDGES 1600000
#define FEATS   256

// ---------------------------------------------------------------------------
// Kernel 1: zero the output accumulator (graph-capture-safe, no memset).
// ---------------------------------------------------------------------------
__global__ __launch_bounds__(256) void gcn_zero(float* __restrict__ out, int n) {
    int i = blockIdx.x * 256 + threadIdx.x;
    if (i < n) out[i] = 0.0f;
}

// ---------------------------------------------------------------------------
// Kernel 2: x = norm * (h @ W^T + b) via V_WMMA_F32_16X16X4_F32.
// One wave computes a 16(nodes) x 16(out-feats) tile. K loops 256 in steps of 4.
//
// Layouts (CDNA5 ISA):
//  A 16x4 f32:  lane l -> M = l%16, VGPR v -> K = v + 2*(l/16)   (contig float2)
//  B 4x16 f32:  lane l -> N = l%16, VGPR v -> K = v + 2*(l/16)   (contig float2
//               of W's row, since B[k][n] = W[n][k])
//  C/D 16x16:   lane l, VGPR v -> M = v + 8*(l/16), N = l%16
// ---------------------------------------------------------------------------
__global__ __launch_bounds__(256) void gcn_gemm_wmma(
    const float* __restrict__ h, const float* __restrict__ norm,
    const float* __restrict__ W, const float* __restrict__ bvec,
    float* __restrict__ x)
{
    const int lane = threadIdx.x & 31;
    const int wave = threadIdx.x >> 5;
    const int hi   = lane >> 4;   // 0 or 1
    const int lo   = lane & 15;

    const int node_base = (blockIdx.x * 8 + wave) * 16;
    const int feat_base = blockIdx.y * 16;

    // wave-uniform exit: EXEC stays all-1s for waves that run WMMA
    if (node_base >= N_NODES) return;

    // A-frag source row (clamped per-lane so EXEC stays full through the loop;
    // with N_NODES % 16 == 0 the clamp is never taken, but keep it generic)
    int am   = node_base + lo;
    int am_c = (am < N_NODES) ? am : (N_NODES - 1);
    const float* __restrict__ hrow = h + (long)am_c * FEATS;

    // B-frag source row: output feature (lo) of W, K-pair contiguous
    const float* __restrict__ wrow = W + (long)(feat_base + lo) * FEATS;

    v8f acc = {0.f, 0.f, 0.f, 0.f, 0.f, 0.f, 0.f, 0.f};

    #pragma unroll 8
    for (int k = 0; k < FEATS; k += 4) {
        v2f a  = *(const v2f*)(hrow + k + 2 * hi);   // h[m][k+2hi .. +1]
        v2f bb = *(const v2f*)(wrow + k + 2 * hi);   // W[n][k+2hi .. +1]
        // 8 args: (neg_a, A, neg_b, B, c_mod, C, reuse_a, reuse_b)
        acc = __builtin_amdgcn_wmma_f32_16x16x4_f32(
            false, a, false, bb, (short)0, acc, false, false);
    }

    const float bias = bvec[feat_base + lo];
    float* __restrict__ xcol = x + (long)node_base * FEATS + feat_base + lo;

    if (node_base + 16 <= N_NODES) {
        // fast path (always taken when N_NODES % 16 == 0): wave-uniform,
        // 8 unconditional stores, no EXEC churn
        #pragma unroll
        for (int v = 0; v < 8; ++v) {
            const int m = v + 8 * hi;
            xcol[(long)m * FEATS] = norm[node_base + m] * (acc[v] + bias);
        }
    } else {
        // cold fallback for a partial tile
        #pragma unroll
        for (int v = 0; v < 8; ++v) {
            const int m = node_base + v + 8 * hi;
            if (m < N_NODES)
                x[(long)m * FEATS + feat_base + lo] = norm[m] * (acc[v] + bias);
        }
    }
}

// ---------------------------------------------------------------------------
// Kernel 3: edge scatter  out[dst] += x[src].  One wave per edge:
// 32 lanes x (2 x float4) cover 256 feats; coalesced gather, L2-resident
// non-returning f32 atomics (global_atomic_add_f32, STOREcnt path).
// ---------------------------------------------------------------------------
__global__ __launch_bounds__(256) void gcn_scatter(
    const float* __restrict__ x, const int* __restrict__ src,
    const int* __restrict__ dst, float* __restrict__ out)
{
    const int lane = threadIdx.x & 31;
    const int wave = threadIdx.x >> 5;
    const long e = (long)blockIdx.x * 8 + wave;
    if (e >= N_EDGES) return;

    const int s = src[e];
    const int d = dst[e];
    const float* __restrict__ xs = x + (long)s * FEATS;
    float* __restrict__ od = out + (long)d * FEATS;

    #pragma unroll
    for (int c = 0; c < 2; ++c) {
        const int f = c * 128 + lane * 4;
        v4f val = *(const v4f*)(xs + f);
        atomicAdd(od + f + 0, val.x);
        atomicAdd(od + f + 1, val.y);
        atomicAdd(od + f + 2, val.z);
        atomicAdd(od + f + 3, val.w);
    }
}

// ---------------------------------------------------------------------------
// Kernel 4: out *= norm (post-aggregation scaling)
// ---------------------------------------------------------------------------
__global__ __launch_bounds__(256) void gcn_postscale(
    float* __restrict__ out, const float* __restrict__ norm)
{
    long i = (long)blockIdx.x * 256 + threadIdx.x;
    if (i < (long)N_NODES * FEATS) out[i] *= norm[i >> 8];  // FEATS == 256
}

// ---------------------------------------------------------------------------
extern "C" void kernel_launch(void* const* d_in, const int* in_sizes, int n_in,
                              void* d_out, int out_size, void* d_ws, size_t ws_size,
                              hipStream_t stream)
{
    (void)in_sizes; (void)n_in; (void)out_size; (void)ws_size;

    const float* h    = (const float*)d_in[0];
    const float* norm = (const float*)d_in[1];
    const float* W    = (const float*)d_in[2];
    const float* bias = (const float*)d_in[3];
    const int*   src  = (const int*)d_in[4];
    const int*   dst  = (const int*)d_in[5];

    float* out = (float*)d_out;
    float* x   = (float*)d_ws;   // staging: 100000 * 256 * 4 B = 102.4 MB

    const int total = N_NODES * FEATS;

    // 1) zero accumulator
    gcn_zero<<<(total + 255) / 256, 256, 0, stream>>>(out, total);

    // 2) x = norm * (h @ W^T + b)   [WMMA f32 16x16x4]
    dim3 g((N_NODES + 127) / 128, FEATS / 16);
    gcn_gemm_wmma<<<g, 256, 0, stream>>>(h, norm, W, bias, x);

    // 3) out[dst] += x[src]
    gcn_scatter<<<(N_EDGES + 7) / 8, 256, 0, stream>>>(x, src, dst, out);

    // 4) out *= norm
    gcn_postscale<<<(total + 255) / 256, 256, 0, stream>>>(out, norm);
}